// AttentionSelf_64123861729632
// MI455X (gfx1250) — compile-verified
//
#include <hip/hip_runtime.h>
#include <hip/hip_bf16.h>

// CDNA5 / gfx1250. out = X @ (X^T @ X) per batch, fp32.
// Matrix op: V_WMMA_F32_16X16X4_F32 (fp32 A/B, fp32 accum) -> same precision
// class as the fp32 reference. X (64MB) + gram (8MB) fit in the 192MB L2, so
// operand gathers are L2 hits after first touch; HBM floor ~5.5us.
// 4x4 WMMA macro-tiles (64x64 per wave): 16 wmma per K-step against 12-16
// b32/b64 loads -> matrix-pipe issue rate is no longer gated by vmem issue.

typedef __attribute__((ext_vector_type(2))) float v2f;
typedef __attribute__((ext_vector_type(8))) float v8f;

#define NB 8
#define NS 4096
#define ND 512

// D = A(16x4) * B(4x16) + C, fp32. Per-lane layouts (wave32):
//   A: a[v] = A[M = lane%16][K = 2*(lane/16) + v]
//   B: b[v] = B[K = 2*(lane/16) + v][N = lane%16]
//   C/D: c[v] = C[M = v + 8*(lane/16)][N = lane%16]
static __device__ __forceinline__ v8f wmma_f32(v2f a, v2f b, v8f c) {
  return __builtin_amdgcn_wmma_f32_16x16x4_f32(
      /*neg_a=*/false, a, /*neg_b=*/false, b,
      /*c_mod=*/(short)0, c, /*reuse_a=*/false, /*reuse_b=*/false);
}

// ---------------------------------------------------------------------------
// Phase 1: gram[b] = X[b]^T @ X[b]   (ND x ND, reduce over NS)
// One wave per 64x64 macro-tile (4x4 WMMA tiles). 64 macro-tiles/batch.
// gram[d,e] = sum_s X[s,d]*X[s,e]  -> A[m,k]=X[s0+k,d0+m], B[k,n]=X[s0+k,e0+n]
// ---------------------------------------------------------------------------
__global__ __launch_bounds__(256) void gram_kernel(const float* __restrict__ X,
                                                   float* __restrict__ gram) {
  const int lane = threadIdx.x & 31;
  const int wave = threadIdx.x >> 5;
  const int w = blockIdx.x * 8 + wave;     // global wave id
  const int b = w >> 6;                    // 64 macro tiles per batch
  const int t = w & 63;
  const int d0 = (t >> 3) << 6;            // macro row (64)
  const int e0 = (t & 7) << 6;             // macro col (64)

  const int m  = lane & 15;                // M / N index within tile
  const int kl = (lane >> 4) << 1;         // K base for lane half (0 or 2)

  const float* Xb = X + (size_t)b * NS * ND;

  v8f acc[4][4] = {};

  for (int s0 = 0; s0 < NS; s0 += 4) {
    const float* r0 = Xb + (size_t)(s0 + kl) * ND;   // row s0+kl
    const float* r1 = r0 + ND;                       // row s0+kl+1
    v2f a[4], bb[4];
#pragma unroll
    for (int i = 0; i < 4; ++i) {
      a[i].x = r0[d0 + 16 * i + m];
      a[i].y = r1[d0 + 16 * i + m];
    }
#pragma unroll
    for (int j = 0; j < 4; ++j) {
      bb[j].x = r0[e0 + 16 * j + m];
      bb[j].y = r1[e0 + 16 * j + m];
    }
#pragma unroll
    for (int i = 0; i < 4; ++i)
#pragma unroll
      for (int j = 0; j < 4; ++j)
        acc[i][j] = wmma_f32(a[i], bb[j], acc[i][j]);
  }

  float* Gb = gram + (size_t)b * ND * ND;
  const int rbase = (lane >> 4) << 3;      // 0 or 8
#pragma unroll
  for (int i = 0; i < 4; ++i)
#pragma unroll
    for (int j = 0; j < 4; ++j)
#pragma unroll
      for (int v = 0; v < 8; ++v)
        Gb[(size_t)(d0 + 16 * i + rbase + v) * ND + e0 + 16 * j + m] = acc[i][j][v];
}

// ---------------------------------------------------------------------------
// Phase 2: out[b] = X[b] @ gram[b]   (NS x ND, reduce over ND)
// One wave per 64x64 macro-tile (4x4 WMMA tiles). 512 macro-tiles/batch.
// A[m,k]=X[s0+m,k0+k] (contiguous float2 per lane), B[k,n]=gram[k0+k,e0+n]
// ---------------------------------------------------------------------------
__global__ __launch_bounds__(256) void out_kernel(const float* __restrict__ X,
                                                  const float* __restrict__ gram,
                                                  float* __restrict__ out) {
  const int lane = threadIdx.x & 31;
  const int wave = threadIdx.x >> 5;
  const int w = blockIdx.x * 8 + wave;
  const int b = w >> 9;                    // 512 macro tiles per batch
  const int t = w & 511;
  const int s0 = (t >> 3) << 6;            // 64 row blocks of 64
  const int e0 = (t & 7) << 6;             // 8 col blocks of 64

  const int m  = lane & 15;
  const int kl = (lane >> 4) << 1;

  const float* Xb = X + (size_t)b * NS * ND;
  const float* Gb = gram + (size_t)b * ND * ND;

  v8f acc[4][4] = {};

  for (int k0 = 0; k0 < ND; k0 += 4) {
    v2f a[4], bb[4];
#pragma unroll
    for (int i = 0; i < 4; ++i)   // 8B-aligned float2 along K
      a[i] = *(const v2f*)(Xb + (size_t)(s0 + 16 * i + m) * ND + k0 + kl);
    const float* g0 = Gb + (size_t)(k0 + kl) * ND;   // row k0+kl of gram
    const float* g1 = g0 + ND;                       // row k0+kl+1
#pragma unroll
    for (int j = 0; j < 4; ++j) {
      bb[j].x = g0[e0 + 16 * j + m];
      bb[j].y = g1[e0 + 16 * j + m];
    }
#pragma unroll
    for (int i = 0; i < 4; ++i)
#pragma unroll
      for (int j = 0; j < 4; ++j)
        acc[i][j] = wmma_f32(a[i], bb[j], acc[i][j]);
  }

  float* Ob = out + (size_t)b * NS * ND;
  const int rbase = (lane >> 4) << 3;
#pragma unroll
  for (int i = 0; i < 4; ++i)
#pragma unroll
    for (int j = 0; j < 4; ++j)
#pragma unroll
      for (int v = 0; v < 8; ++v)
        Ob[(size_t)(s0 + 16 * i + rbase + v) * ND + e0 + 16 * j + m] = acc[i][j][v];
}

extern "C" void kernel_launch(void* const* d_in, const int* in_sizes, int n_in,
                              void* d_out, int out_size, void* d_ws, size_t ws_size,
                              hipStream_t stream) {
  const float* X = (const float*)d_in[0];          // [8, 4096, 512] fp32
  float* out = (float*)d_out;                      // [8, 4096, 512] fp32
  float* gram = (float*)d_ws;                      // [8, 512, 512] fp32 = 8 MB scratch

  // Phase 1: 8 batches * 64 macro-tiles = 512 waves / 8 waves-per-block
  gram_kernel<<<64, 256, 0, stream>>>(X, gram);
  // Phase 2: 8 batches * 512 macro-tiles = 4096 waves / 8 waves-per-block
  out_kernel<<<512, 256, 0, stream>>>(X, gram, out);
}